// CumulativeFlattenedLinear_19490561590059
// MI455X (gfx1250) — compile-verified
//
#include <hip/hip_runtime.h>
#include <hip/hip_bf16.h>
#include <stddef.h>

typedef float v2f __attribute__((ext_vector_type(2)));
typedef float v4f __attribute__((ext_vector_type(4)));
typedef float v8f __attribute__((ext_vector_type(8)));

#define B_      8
#define C_      16
#define T_      131072
#define O_      16
#define NKEEP   48
#define NDISC   16
#define S_      64
#define G_      (T_ / S_)              /* 2048 windows */
#define MTOT    (B_ * G_)              /* 16384 (b,g) rows */
#define TILES   (MTOT / 16)            /* 1024 row-tiles, one per BLOCK */
#define WPB     4                      /* waves per block: 4 x 16 timesteps */
#define CHUNK   16                     /* timesteps per wave */
#define THREADS (WPB * 32)

// Block = 4 waves, one 16-row (b,g) tile, one 64-step window.
// Wave w owns timesteps [16w, 16w+16): per step, 4 chained
// V_WMMA_F32_16X16X4_F32 (K=C=16) with the accumulator carried across steps
// -> wave-local causal cumsum, every per-step 16x16 tile kept in registers.
// Chunk totals are exchanged through LDS (block-level exclusive scan), then
// prefix+bias is added at store time. 4096 waves total for HBM latency hiding.
__global__ __launch_bounds__(THREADS)
void cfl_wmma_kernel(const float* __restrict__ x,
                     const float* __restrict__ weight,
                     const float* __restrict__ bias,
                     float* __restrict__ out) {
  // Expanded per-timestep weights in WMMA-B layout:
  // wlds[s][kc][half][o] = { W[s][c0][o], W[s][c0+1][o] },  c0 = 4*kc + 2*half
  __shared__ v2f wlds[S_][4][2][16];
  __shared__ float part[WPB - 1][16][16];   // chunk-total tiles of waves 0..2

  const int tid = threadIdx.x;
  for (int e = tid; e < S_ * 4 * 2 * 16; e += THREADS) {
    const int o    = e & 15;
    const int hf   = (e >> 4) & 1;
    const int kc   = (e >> 5) & 3;
    const int s    = e >> 7;
    const int c0   = kc * 4 + hf * 2;
    const int t    = s - NDISC;               // discarded slots -> zero block
    v2f wv;
    wv.x = (t >= 0) ? weight[o * (C_ * NKEEP) + (c0    ) * NKEEP + t] : 0.0f;
    wv.y = (t >= 0) ? weight[o * (C_ * NKEEP) + (c0 + 1) * NKEEP + t] : 0.0f;
    wlds[s][kc][hf][o] = wv;
  }
  __syncthreads();

  const int wave = tid >> 5;
  const int lane = tid & 31;
  const int m0   = blockIdx.x * 16;
  const int b    = m0 / G_;                   // 16 | G_: tile shares b
  const int g0   = m0 % G_;
  const int sb   = wave * CHUNK;              // this wave's timestep base

  const int low  = lane & 15;                 // A: M row; B/C/D: N col
  const int half = lane >> 4;
  const int hi8  = half << 3;                 // C/D rows 8..15 live here

  // Per-lane A pointers: x[b, c, (g0+low)*64 + sb + s]; c = 4*kc + 2*half (+1)
  const float* aptr[4];
#pragma unroll
  for (int kc = 0; kc < 4; ++kc) {
    const int c = kc * 4 + half * 2;
    aptr[kc] = x + ((size_t)b * C_ + c) * T_ + (size_t)(g0 + low) * S_ + sb;
  }

  // ---- Phase 1: wave-local cumsum, all 16 tiles held in registers --------
  v8f acc = { 0.f, 0.f, 0.f, 0.f, 0.f, 0.f, 0.f, 0.f };
  v8f zb[CHUNK];
#pragma unroll
  for (int s4 = 0; s4 < CHUNK; s4 += 4) {
    v4f alo[4], ahi[4];
#pragma unroll
    for (int kc = 0; kc < 4; ++kc) {
      alo[kc] = *(const v4f*)(aptr[kc] + s4);
      ahi[kc] = *(const v4f*)(aptr[kc] + T_ + s4);
    }
    if (s4 + 4 < CHUNK) {                     // uniform per wave
      __builtin_prefetch(aptr[0] + s4 + 4, 0, 1);
      __builtin_prefetch(aptr[3] + T_ + s4 + 4, 0, 1);
    }
#pragma unroll
    for (int u = 0; u < 4; ++u) {
      const int sl = s4 + u;
      const int s  = sb + sl;
#pragma unroll
      for (int kc = 0; kc < 4; ++kc) {
        v2f av = { alo[kc][u], ahi[kc][u] };  // A VGPR0/1 = K=c0 / K=c0+1
        v2f bw = wlds[s][kc][half][low];      // ds_load_b64, conflict-free
        acc = __builtin_amdgcn_wmma_f32_16x16x4_f32(
                  false, av, false, bw, (short)0, acc, false, false);
      }
      zb[sl] = acc;                           // running cumsum tile snapshot
    }
  }

  // ---- Block-level exclusive scan of chunk totals ------------------------
  if (wave < WPB - 1) {
#pragma unroll
    for (int j = 0; j < 8; ++j) part[wave][j + hi8][low] = acc[j];
  }
  __syncthreads();

  const float bv = bias[low];
  v8f base = { bv, bv, bv, bv, bv, bv, bv, bv };
  for (int pw = 0; pw < WPB - 1; ++pw) {
    if (wave > pw) {
#pragma unroll
      for (int j = 0; j < 8; ++j) base[j] += part[pw][j + hi8][low];
    }
  }

  // ---- Phase 2: add prefix + bias, store b128 per 4 timesteps ------------
  float* ob = out + ((size_t)b * O_ + low) * T_ + sb;
#pragma unroll
  for (int s4 = 0; s4 < CHUNK; s4 += 4) {
#pragma unroll
    for (int j = 0; j < 8; ++j) {
      v4f zz = { zb[s4 + 0][j] + base[j], zb[s4 + 1][j] + base[j],
                 zb[s4 + 2][j] + base[j], zb[s4 + 3][j] + base[j] };
      *(v4f*)(ob + (size_t)(g0 + j + hi8) * S_ + s4) = zz;
    }
  }
}

extern "C" void kernel_launch(void* const* d_in, const int* in_sizes, int n_in,
                              void* d_out, int out_size, void* d_ws, size_t ws_size,
                              hipStream_t stream) {
  (void)in_sizes; (void)n_in; (void)d_ws; (void)ws_size; (void)out_size;
  const float* x      = (const float*)d_in[0];
  const float* weight = (const float*)d_in[1];
  const float* bias   = (const float*)d_in[2];
  float* out          = (float*)d_out;
  dim3 grid(TILES);          // 1024 blocks x 128 threads = 4096 waves
  dim3 block(THREADS);
  cfl_wmma_kernel<<<grid, block, 0, stream>>>(x, weight, bias, out);
}